// LayoutModel_73065983640002
// MI455X (gfx1250) — compile-verified
//
#include <hip/hip_runtime.h>
#include <hip/hip_bf16.h>

// ---------------------------------------------------------------------------
// TpuGraphs-style layout GNN for MI455X (gfx1250, wave32).
// Dense matmuls (output width 64) run on V_WMMA_F32_16X16X4_F32 with two
// independent accumulator chains (agg@Wl and x@Wr) to avoid serial WMMA
// dependencies. Mean-aggregation uses fp32 hardware atomics with float4
// source loads; 1/deg scaling + M-tail masking are folded into the WMMA
// A-operand scale (branchless -> no exec-mask churn in the inner loop).
// ---------------------------------------------------------------------------

#define NN   100000
#define EE   1000000
#define CC   48
#define NCC  5000
#define ECC  20000

typedef float v2f __attribute__((ext_vector_type(2)));
typedef float v4f __attribute__((ext_vector_type(4)));
typedef float v8f __attribute__((ext_vector_type(8)));

__device__ __forceinline__ float leaky(float v) { return v >= 0.0f ? v : 0.01f * v; }

__device__ __forceinline__ void atomAddF(float* p, float v) {
    unsafeAtomicAdd(p, v);   // global_atomic_add_f32 on gfx1250
}

// -------------------------------- utility ----------------------------------
__global__ void k_zero(float* __restrict__ p, int n) {
    int t = blockIdx.x * blockDim.x + threadIdx.x;
    if (t < n) p[t] = 0.0f;
}

// x0[i, 0:140] = node_feat[i,:] ; x0[i, 140:172] = emb[opcode[i],:]
__global__ void k_concat_embed(const float* __restrict__ nf, const int* __restrict__ opc,
                               const float* __restrict__ emb, float* __restrict__ x0, int n) {
    int t = blockIdx.x * blockDim.x + threadIdx.x;
    if (t >= n * 172) return;
    int i = t / 172, f = t % 172;
    x0[t] = (f < 140) ? nf[i * 140 + f] : emb[opc[i] * 32 + (f - 140)];
}

// deg[dst] += 1 per edge
__global__ void k_deg(const int* __restrict__ edges, int E, float* __restrict__ deg) {
    int e = blockIdx.x * blockDim.x + threadIdx.x;
    if (e < E) atomAddF(&deg[edges[E + e]], 1.0f);
}

// batched segment-sum: agg[b,dst,f:f+4] += x[b,src,f:f+4]  (din % 4 == 0)
__global__ void k_scatter4(const float* __restrict__ x, const int* __restrict__ edges,
                           int E, int nNodes, int din, int B, float* __restrict__ agg) {
    int t = blockIdx.x * blockDim.x + threadIdx.x;
    int dq = din >> 2;
    int total = B * E * dq;
    if (t >= total) return;
    int q = t % dq;
    int r = t / dq;
    int e = r % E;
    int b = r / E;
    int src = edges[e];
    int dst = edges[E + e];
    const v4f v = *(const v4f*)(x + ((size_t)b * nNodes + src) * din + q * 4);
    float* d = agg + ((size_t)b * nNodes + dst) * din + q * 4;
    atomAddF(d + 0, v[0]);
    atomAddF(d + 1, v[1]);
    atomAddF(d + 2, v[2]);
    atomAddF(d + 3, v[3]);
}

// ------------------------- SAGE GEMM via WMMA ------------------------------
// out[m,0:64] = leaky( (agg[m,:]/max(deg,1)) @ Wl + bl + x[m,:] @ Wr )
// block = 256 threads = 8 waves; each wave computes one 16x16 f32 tile.
// K must be a multiple of 4 (172 / 192 / 64 all qualify).
__global__ void k_sage_gemm(const float* __restrict__ Agg, const float* __restrict__ X,
                            const float* __restrict__ Wl, const float* __restrict__ bl,
                            const float* __restrict__ Wr, const float* __restrict__ deg,
                            int degMod, float* __restrict__ out, int M, int K) {
    const int lane = threadIdx.x & 31;
    const int wave = threadIdx.x >> 5;            // 0..7
    const int half = lane >> 4;                    // 0/1 -> K pair select
    const int l    = lane & 15;
    const int m0 = blockIdx.x * 32 + (wave >> 2) * 16;
    const int n0 = (wave & 3) * 16;
    const int rA = m0 + l;
    const float vmask = (rA < M) ? 1.0f : 0.0f;    // branchless M-tail mask
    const int rC = (rA < M) ? rA : (M - 1);        // clamped, always-safe address
    const int col = n0 + l;

    float aggScale = vmask;
    if (deg) aggScale = vmask / fmaxf(deg[rC % degMod], 1.0f);

    v8f cl, cr;                                    // two independent WMMA chains
    const float bias = bl ? bl[col] : 0.0f;
#pragma unroll
    for (int r = 0; r < 8; ++r) { cl[r] = bias; cr[r] = 0.0f; }

    // A layout (16x4 f32): lanes 0-15 -> K={k,k+1}, lanes 16-31 -> K={k+2,k+3}
    // B layout (4x16 f32): v0 holds K=k (lo lanes) / k+2 (hi lanes), v1: k+1 / k+3
    const float* arow = Agg + (size_t)rC * K + 2 * half;
    const float* xrow = X   + (size_t)rC * K + 2 * half;
    const float* wlp  = Wl + (2 * half) * 64 + col;
    const float* wrp  = Wr + (2 * half) * 64 + col;

#pragma unroll 2
    for (int k = 0; k < K; k += 4) {
        v2f aa = *(const v2f*)(arow + k);          // global_load_b64
        v2f ax = *(const v2f*)(xrow + k);
        v2f bwl, bwr;
        bwl[0] = wlp[(size_t)k * 64];
        bwl[1] = wlp[(size_t)(k + 1) * 64];
        bwr[0] = wrp[(size_t)k * 64];
        bwr[1] = wrp[(size_t)(k + 1) * 64];
        aa[0] *= aggScale; aa[1] *= aggScale;
        ax[0] *= vmask;    ax[1] *= vmask;
        cl = __builtin_amdgcn_wmma_f32_16x16x4_f32(false, aa, false, bwl, (short)0, cl, false, false);
        cr = __builtin_amdgcn_wmma_f32_16x16x4_f32(false, ax, false, bwr, (short)0, cr, false, false);
    }
#pragma unroll
    for (int r = 0; r < 8; ++r) {                  // C layout: VGPR r -> M = r + 8*half
        int m = m0 + r + half * 8;
        if (m < M) out[(size_t)m * 64 + n0 + l] = leaky(cl[r] + cr[r]);
    }
}

// nbr[i,:] = agg[ids[i],:] / max(deg[ids[i]],1) ; xs[i,:] = x[ids[i],:]
__global__ void k_scale_gather(const float* __restrict__ agg, const float* __restrict__ deg,
                               const int* __restrict__ ids, float* __restrict__ outNbr,
                               const float* __restrict__ x, float* __restrict__ outXs, int nc) {
    int t = blockIdx.x * blockDim.x + threadIdx.x;
    if (t >= nc * 64) return;
    int i = t / 64, f = t % 64;
    int g = ids[i];
    float s = 1.0f / fmaxf(deg[g], 1.0f);
    outNbr[t] = agg[(size_t)g * 64 + f] * s;
    outXs[t]  = x[(size_t)g * 64 + f];
}

// merged[c,i,0:64]=nbr ; [64:128]=xs ; [128:192]=leaky(cfgfeat@prjW+prjb) ; L2-normalize row
__global__ void k_build_merged(const float* __restrict__ nbr, const float* __restrict__ xs,
                               const float* __restrict__ ncf, const float* __restrict__ prjW,
                               const float* __restrict__ prjb, float* __restrict__ merged, int nc) {
    int row = blockIdx.x;                          // c*NC + i
    int i = row % nc;
    int t = threadIdx.x;                           // 256 threads, 192 produce data
    __shared__ float sv[192];
    __shared__ float red[256];
    float v = 0.0f;
    if (t < 64) {
        v = nbr[(size_t)i * 64 + t];
    } else if (t < 128) {
        v = xs[(size_t)i * 64 + (t - 64)];
    } else if (t < 192) {
        int f = t - 128;
        const float* cfr = ncf + (size_t)row * 18;
        float acc = prjb[f];
#pragma unroll
        for (int k = 0; k < 18; ++k) acc += cfr[k] * prjW[k * 64 + f];
        v = leaky(acc);
    }
    if (t < 192) sv[t] = v;
    red[t] = (t < 192) ? v * v : 0.0f;
    __syncthreads();
    for (int s = 128; s > 0; s >>= 1) {
        if (t < s) red[t] += red[t + s];
        __syncthreads();
    }
    float scale = 1.0f / fmaxf(sqrtf(red[0]), 1e-12f);
    if (t < 192) merged[(size_t)row * 192 + t] = sv[t] * scale;
}

// pooled[c,f] = mean_i y[c,i,f]
__global__ void k_pool_mean(const float* __restrict__ y, float* __restrict__ pooled, int nc) {
    int c = blockIdx.x;
    int t = threadIdx.x;                           // 256
    int f = t & 63, chunk = t >> 6;                // 4 chunks per feature
    float acc = 0.0f;
    for (int i = chunk; i < nc; i += 4) acc += y[((size_t)c * nc + i) * 64 + f];
    __shared__ float red[256];
    red[t] = acc;
    __syncthreads();
    if (chunk == 0)
        pooled[c * 64 + f] = (red[f] + red[f + 64] + red[f + 128] + red[f + 192]) / (float)nc;
}

// small dense: out[m,n] = (leaky?)( sum_k in[m,k]*W[k,n] )
__global__ void k_dense(const float* __restrict__ in, const float* __restrict__ W,
                        float* __restrict__ out, int M, int K, int Nn, int applyLeaky) {
    int t = blockIdx.x * blockDim.x + threadIdx.x;
    if (t >= M * Nn) return;
    int m = t / Nn, n = t % Nn;
    float acc = 0.0f;
    for (int k = 0; k < K; ++k) acc += in[(size_t)m * K + k] * W[k * Nn + n];
    out[t] = applyLeaky ? leaky(acc) : acc;
}

// ---------------------------------------------------------------------------
static inline int blocks(long long total, int bs) { return (int)((total + bs - 1) / bs); }

extern "C" void kernel_launch(void* const* d_in, const int* in_sizes, int n_in,
                              void* d_out, int out_size, void* d_ws, size_t ws_size,
                              hipStream_t stream) {
    (void)in_sizes; (void)n_in; (void)out_size; (void)ws_size;

    const float* node_feat = (const float*)d_in[0];
    const int*   opcode    = (const int*)  d_in[1];
    const int*   edges     = (const int*)  d_in[2];     // [2,E] row-major: src then dst
    const float* cfg_feat  = (const float*)d_in[3];     // [C,NC,18]
    const int*   cfg_ids   = (const int*)  d_in[4];     // [NC]
    const int*   cedges    = (const int*)  d_in[5];     // [2,EC]

    // params flattened in setup_inputs() insertion order
    const float* emb = (const float*)d_in[6];
    const float *nWl[3], *nbl[3], *nWr[3], *cnWl[3], *cnbl[3], *cnWr[3], *cgWl[3], *cgbl[3], *cgWr[3];
    for (int L = 0; L < 3; ++L) {
        nWl[L]  = (const float*)d_in[7  + 3 * L]; nbl[L]  = (const float*)d_in[8  + 3 * L]; nWr[L]  = (const float*)d_in[9  + 3 * L];
        cnWl[L] = (const float*)d_in[16 + 3 * L]; cnbl[L] = (const float*)d_in[17 + 3 * L]; cnWr[L] = (const float*)d_in[18 + 3 * L];
        cgWl[L] = (const float*)d_in[25 + 3 * L]; cgbl[L] = (const float*)d_in[26 + 3 * L]; cgWr[L] = (const float*)d_in[27 + 3 * L];
    }
    const float* prjW = (const float*)d_in[34];
    const float* prjb = (const float*)d_in[35];
    const float* dW1  = (const float*)d_in[36];
    const float* dW2  = (const float*)d_in[37];
    const float* dW3  = (const float*)d_in[38];
    float* outp = (float*)d_out;

    // ---- workspace layout (floats; every offset is an even float count) ----
    float* w = (float*)d_ws;
    const size_t N172 = (size_t)NN * 172, N64 = (size_t)NN * 64;
    const size_t NC64 = (size_t)NCC * 64;
    const size_t CN192 = (size_t)CC * NCC * 192, CN64 = (size_t)CC * NCC * 64;
    size_t o = 0;
    float* x0    = w + o; o += N172;
    float* agg   = w + o; o += N172;    // node agg (sized for widest layer)
    float* xA    = w + o; o += N64;
    float* xB    = w + o; o += N64;
    float* degN  = w + o; o += NN;
    float* aggS  = w + o; o += NC64;    // config-graph agg (64-wide)
    float* nbrA  = w + o; o += NC64;
    float* nbrB  = w + o; o += NC64;
    float* xs    = w + o; o += NC64;
    float* degC  = w + o; o += NCC;
    float* cfgX  = w + o; o += CN192;   // merged; later reused as 64-wide ping buffer
    float* cfgAgg= w + o; o += CN192;
    float* cfgY  = w + o; o += CN64;
    float* pooled= w + o; o += (size_t)CC * 64;
    float* h1    = w + o; o += (size_t)CC * 64;
    float* h2    = w + o; o += (size_t)CC * 64;

    const int BS = 256;

    // ---- node graph degrees ----
    k_zero<<<blocks(NN, BS), BS, 0, stream>>>(degN, NN);
    k_deg<<<blocks(EE, BS), BS, 0, stream>>>(edges, EE, degN);

    // ---- x0 = concat(node_feat, emb[opcode]) ----
    k_concat_embed<<<blocks((long long)NN * 172, BS), BS, 0, stream>>>(node_feat, opcode, emb, x0, NN);

    // ---- node_gnn: 3 SAGE layers ----
    {   // L0: 172 -> 64
        k_zero<<<blocks(N172, BS), BS, 0, stream>>>(agg, (int)N172);
        k_scatter4<<<blocks((long long)EE * 43, BS), BS, 0, stream>>>(x0, edges, EE, NN, 172, 1, agg);
        k_sage_gemm<<<(NN + 31) / 32, BS, 0, stream>>>(agg, x0, nWl[0], nbl[0], nWr[0], degN, NN, xA, NN, 172);
    }
    {   // L1: 64 -> 64
        k_zero<<<blocks(N64, BS), BS, 0, stream>>>(agg, (int)N64);
        k_scatter4<<<blocks((long long)EE * 16, BS), BS, 0, stream>>>(xA, edges, EE, NN, 64, 1, agg);
        k_sage_gemm<<<(NN + 31) / 32, BS, 0, stream>>>(agg, xA, nWl[1], nbl[1], nWr[1], degN, NN, xB, NN, 64);
    }
    {   // L2: 64 -> 64
        k_zero<<<blocks(N64, BS), BS, 0, stream>>>(agg, (int)N64);
        k_scatter4<<<blocks((long long)EE * 16, BS), BS, 0, stream>>>(xB, edges, EE, NN, 64, 1, agg);
        k_sage_gemm<<<(NN + 31) / 32, BS, 0, stream>>>(agg, xB, nWl[2], nbl[2], nWr[2], degN, NN, xA, NN, 64);
    }

    // ---- nbr = mean_agg(x_final)[ids] ; xs = x_final[ids] ----
    k_zero<<<blocks(N64, BS), BS, 0, stream>>>(agg, (int)N64);
    k_scatter4<<<blocks((long long)EE * 16, BS), BS, 0, stream>>>(xA, edges, EE, NN, 64, 1, agg);
    k_scale_gather<<<blocks((long long)NCC * 64, BS), BS, 0, stream>>>(agg, degN, cfg_ids, nbrA, xA, xs, NCC);

    // ---- config graph degrees ----
    k_zero<<<blocks(NCC, BS), BS, 0, stream>>>(degC, NCC);
    k_deg<<<blocks(ECC, BS), BS, 0, stream>>>(cedges, ECC, degC);

    // ---- cfg_nbr_gnn: 3 SAGE layers on [NC,64] ----
    const float* nin[3] = {nbrA, nbrB, nbrA};
    float* nout[3] = {nbrB, nbrA, nbrB};
    for (int L = 0; L < 3; ++L) {
        k_zero<<<blocks(NC64, BS), BS, 0, stream>>>(aggS, (int)NC64);
        k_scatter4<<<blocks((long long)ECC * 16, BS), BS, 0, stream>>>(nin[L], cedges, ECC, NCC, 64, 1, aggS);
        k_sage_gemm<<<(NCC + 31) / 32, BS, 0, stream>>>(aggS, nin[L], cnWl[L], cnbl[L], cnWr[L],
                                                        degC, NCC, nout[L], NCC, 64);
    }

    // ---- merged = normalize(concat(nbr, xs, leaky(cfg_feat@prjW+prjb))) ----
    k_build_merged<<<CC * NCC, BS, 0, stream>>>(nbrB, xs, cfg_feat, prjW, prjb, cfgX, NCC);

    // ---- cfg_gnn: 3 batched SAGE layers over C configs ----
    {   // L0: 192 -> 64
        k_zero<<<blocks(CN192, BS), BS, 0, stream>>>(cfgAgg, (int)CN192);
        k_scatter4<<<blocks((long long)CC * ECC * 48, BS), BS, 0, stream>>>(cfgX, cedges, ECC, NCC, 192, CC, cfgAgg);
        k_sage_gemm<<<(CC * NCC + 31) / 32, BS, 0, stream>>>(cfgAgg, cfgX, cgWl[0], cgbl[0], cgWr[0],
                                                             degC, NCC, cfgY, CC * NCC, 192);
    }
    {   // L1: 64 -> 64  (output into reused cfgX region)
        k_zero<<<blocks(CN64, BS), BS, 0, stream>>>(cfgAgg, (int)CN64);
        k_scatter4<<<blocks((long long)CC * ECC * 16, BS), BS, 0, stream>>>(cfgY, cedges, ECC, NCC, 64, CC, cfgAgg);
        k_sage_gemm<<<(CC * NCC + 31) / 32, BS, 0, stream>>>(cfgAgg, cfgY, cgWl[1], cgbl[1], cgWr[1],
                                                             degC, NCC, cfgX, CC * NCC, 64);
    }
    {   // L2: 64 -> 64
        k_zero<<<blocks(CN64, BS), BS, 0, stream>>>(cfgAgg, (int)CN64);
        k_scatter4<<<blocks((long long)CC * ECC * 16, BS), BS, 0, stream>>>(cfgX, cedges, ECC, NCC, 64, CC, cfgAgg);
        k_sage_gemm<<<(CC * NCC + 31) / 32, BS, 0, stream>>>(cfgAgg, cfgX, cgWl[2], cgbl[2], cgWr[2],
                                                             degC, NCC, cfgY, CC * NCC, 64);
    }

    // ---- pool + MLP head ----
    k_pool_mean<<<CC, BS, 0, stream>>>(cfgY, pooled, NCC);
    k_dense<<<blocks(CC * 64, BS), BS, 0, stream>>>(pooled, dW1, h1, CC, 64, 64, 1);
    k_dense<<<blocks(CC * 64, BS), BS, 0, stream>>>(h1,     dW2, h2, CC, 64, 64, 1);
    k_dense<<<blocks(CC, BS),      BS, 0, stream>>>(h2,     dW3, outp, CC, 64, 1, 0);
}